// SelectiveScan_79182017069612
// MI455X (gfx1250) — compile-verified
//
#include <hip/hip_runtime.h>
#include <cstdint>
#include <cstddef>

// Problem constants (match reference: B=8, K=4096, D=1024, fp32)
#define B_   8
#define K_   4096
#define D_   1024
#define NC   32            // number of chunks along K
#define LCH  (K_ / NC)     // 128 timesteps per chunk (128 == 2^7)
#define TPB  256           // threads per block == D/4 (each thread owns 4 channels)
#define NBUF 8             // LDS ring slots (power of two)
#define OUTS 6             // async loads kept in flight (< NBUF for slot-reuse slack)

typedef float vf4 __attribute__((ext_vector_type(4)));  // for nontemporal b128 stores

// ---- CDNA5 async global->LDS helpers (gfx1250) ------------------------------

__device__ __forceinline__ void async_load16(uint32_t lds_byte_addr,
                                             int gbyte_off,
                                             const float* sbase) {
  // GVS form: global address is the 64-bit SGPR base plus the signed 32-bit VGPR offset; data lands at the LDS address in the first VGPR
  asm volatile("global_load_async_to_lds_b128 %0, %1, %2"
               :
               : "v"(lds_byte_addr), "v"(gbyte_off), "s"(sbase)
               : "memory");
}

template <int N>
__device__ __forceinline__ void wait_async() {
  asm volatile("s_wait_asynccnt %0" : : "n"(N) : "memory");
}

// Non-temporal 16B store: y is write-once, never re-read by these kernels.
// TH_STORE_NT keeps x resident in the 192MB L2 for the pass-3 re-read.
__device__ __forceinline__ void store_nt16(float* p, float x, float y,
                                           float z, float w) {
  vf4 v; v.x = x; v.y = y; v.z = z; v.w = w;
  __builtin_nontemporal_store(v, reinterpret_cast<vf4*>(p));
}

// ---- Pass 1 / Pass 3: chunk scan (WRITE_Y=false -> emit C; true -> emit y) --

template <bool WRITE_Y>
__global__ __launch_bounds__(TPB) void scan_pass(
    const float* __restrict__ x,
    const float* __restrict__ alpha,
    const float* __restrict__ beta,
    const float* __restrict__ gamma,
    const float* __restrict__ delta,
    const float* __restrict__ Sin,   // chunk entry states (pass 3)
    float* __restrict__ Cout,        // chunk terminal states (pass 1)
    float* __restrict__ y)
{
  __shared__ float4 buf[NBUF][TPB];          // 32 KB ring, 16B slot per thread
  const int t  = threadIdx.x;
  const int c  = blockIdx.x;                 // chunk index
  const int b  = blockIdx.y;                 // batch index
  const int d4 = t << 2;                     // first of 4 owned channels

  const size_t base_e = ((size_t)b * K_ + (size_t)c * LCH) * D_;
  const float* gbase  = x + base_e;

  const float4 al = *reinterpret_cast<const float4*>(alpha + d4);
  const float4 be = *reinterpret_cast<const float4*>(beta  + d4);
  const float a0 = 1.0f / (1.0f + __expf(-al.x));
  const float a1 = 1.0f / (1.0f + __expf(-al.y));
  const float a2 = 1.0f / (1.0f + __expf(-al.z));
  const float a3 = 1.0f / (1.0f + __expf(-al.w));

  float4 ga = make_float4(0.f, 0.f, 0.f, 0.f);
  float4 de = make_float4(0.f, 0.f, 0.f, 0.f);
  float s0 = 0.f, s1 = 0.f, s2 = 0.f, s3 = 0.f;
  if (WRITE_Y) {
    ga = *reinterpret_cast<const float4*>(gamma + d4);
    de = *reinterpret_cast<const float4*>(delta + d4);
    const float4 sv =
        *reinterpret_cast<const float4*>(Sin + ((size_t)(b * NC + c)) * D_ + d4);
    s0 = sv.x; s1 = sv.y; s2 = sv.z; s3 = sv.w;
  }

  const uint32_t lds0 = (uint32_t)(uintptr_t)(&buf[0][t]);
  constexpr int SLOT_BYTES = TPB * 16;

  // Prologue: fill the pipeline with OUTS rows.
#pragma unroll
  for (int i = 0; i < OUTS; ++i)
    async_load16(lds0 + i * SLOT_BYTES, (i * D_ + d4) * 4, gbase);

  // Main loop: retire oldest row, refill, advance recurrence.
  for (int k = 0; k < LCH - OUTS; ++k) {
    wait_async<OUTS - 1>();                  // oldest in-flight row is done
    const float4 xv = buf[k & (NBUF - 1)][t];
    async_load16(lds0 + ((k + OUTS) & (NBUF - 1)) * SLOT_BYTES,
                 ((k + OUTS) * D_ + d4) * 4, gbase);
    s0 = fmaf(a0, s0, be.x * xv.x);
    s1 = fmaf(a1, s1, be.y * xv.y);
    s2 = fmaf(a2, s2, be.z * xv.z);
    s3 = fmaf(a3, s3, be.w * xv.w);
    if (WRITE_Y) {
      store_nt16(y + base_e + (size_t)k * D_ + d4,
                 fmaf(ga.x, s0, de.x * xv.x),
                 fmaf(ga.y, s1, de.y * xv.y),
                 fmaf(ga.z, s2, de.z * xv.z),
                 fmaf(ga.w, s3, de.w * xv.w));
    }
  }

  // Epilogue: everything outstanding is complete; drain the ring.
  wait_async<0>();
#pragma unroll
  for (int k = LCH - OUTS; k < LCH; ++k) {
    const float4 xv = buf[k & (NBUF - 1)][t];
    s0 = fmaf(a0, s0, be.x * xv.x);
    s1 = fmaf(a1, s1, be.y * xv.y);
    s2 = fmaf(a2, s2, be.z * xv.z);
    s3 = fmaf(a3, s3, be.w * xv.w);
    if (WRITE_Y) {
      store_nt16(y + base_e + (size_t)k * D_ + d4,
                 fmaf(ga.x, s0, de.x * xv.x),
                 fmaf(ga.y, s1, de.y * xv.y),
                 fmaf(ga.z, s2, de.z * xv.z),
                 fmaf(ga.w, s3, de.w * xv.w));
    }
  }

  if (!WRITE_Y) {
    *reinterpret_cast<float4*>(Cout + ((size_t)(b * NC + c)) * D_ + d4) =
        make_float4(s0, s1, s2, s3);
  }
}

// ---- Pass 2: exclusive chunk-prefix states: S_j = a^L * S_{j-1} + C_{j-1} ---

__global__ __launch_bounds__(256) void prefix_pass(
    const float* __restrict__ alpha,
    const float* __restrict__ C,
    float* __restrict__ S)
{
  const int idx = blockIdx.x * blockDim.x + threadIdx.x;  // 0 .. B*(D/4)-1
  if (idx >= B_ * (D_ / 4)) return;
  const int b  = idx / (D_ / 4);
  const int d4 = (idx % (D_ / 4)) * 4;

  const float4 al = *reinterpret_cast<const float4*>(alpha + d4);
  float p0 = 1.0f / (1.0f + __expf(-al.x));
  float p1 = 1.0f / (1.0f + __expf(-al.y));
  float p2 = 1.0f / (1.0f + __expf(-al.z));
  float p3 = 1.0f / (1.0f + __expf(-al.w));
  // a^L with L = 128 = 2^7 via 7 squarings
#pragma unroll
  for (int i = 0; i < 7; ++i) { p0 *= p0; p1 *= p1; p2 *= p2; p3 *= p3; }

  float s0 = 0.f, s1 = 0.f, s2 = 0.f, s3 = 0.f;
#pragma unroll
  for (int j = 0; j < NC; ++j) {
    const size_t o = ((size_t)(b * NC + j)) * D_ + d4;
    *reinterpret_cast<float4*>(S + o) = make_float4(s0, s1, s2, s3);
    const float4 cv = *reinterpret_cast<const float4*>(C + o);
    s0 = fmaf(p0, s0, cv.x);
    s1 = fmaf(p1, s1, cv.y);
    s2 = fmaf(p2, s2, cv.z);
    s3 = fmaf(p3, s3, cv.w);
  }
}

// ---- Launch -----------------------------------------------------------------

extern "C" void kernel_launch(void* const* d_in, const int* in_sizes, int n_in,
                              void* d_out, int out_size, void* d_ws, size_t ws_size,
                              hipStream_t stream) {
  (void)in_sizes; (void)n_in; (void)out_size; (void)ws_size;

  const float* x     = (const float*)d_in[0];
  const float* alpha = (const float*)d_in[1];
  const float* beta  = (const float*)d_in[2];
  const float* gamma = (const float*)d_in[3];
  const float* delta = (const float*)d_in[4];
  float* y = (float*)d_out;

  float* C = (float*)d_ws;                       // [B, NC, D] chunk terminal states
  float* S = C + (size_t)B_ * NC * D_;           // [B, NC, D] chunk entry states

  dim3 grid1(NC, B_);

  // Pass 1: per-chunk local scan (s starts at 0) -> C (x pulled into L2, RT)
  scan_pass<false><<<grid1, TPB, 0, stream>>>(x, alpha, beta, gamma, delta,
                                              nullptr, C, nullptr);
  // Pass 2: sequential combine across the 32 chunks -> S (exclusive prefixes)
  prefix_pass<<<(B_ * (D_ / 4) + 255) / 256, 256, 0, stream>>>(alpha, C, S);
  // Pass 3: exact re-scan from S (x re-read hits L2), emit y with NT stores
  scan_pass<true><<<grid1, TPB, 0, stream>>>(x, alpha, beta, gamma, delta,
                                             S, nullptr, y);
}